// Cluster_Level_GCN_6305011990541
// MI455X (gfx1250) — compile-verified
//
#include <hip/hip_runtime.h>
#include <math.h>

typedef float v2f __attribute__((ext_vector_type(2)));
typedef float v8f __attribute__((ext_vector_type(8)));

// ---------------- WMMA helpers (fp32 16x16x4) ----------------
__device__ __forceinline__ v8f wmma_f32(v2f a, v2f b, v8f c) {
  // 8 args: (neg_a, A, neg_b, B, c_mod, C, reuse_a, reuse_b)
  return __builtin_amdgcn_wmma_f32_16x16x4_f32(false, a, false, b, (short)0, c,
                                               false, false);
}

// A-operand fragment from row-major [rows][ld] storage.
// lane<16: m=lane, k=k0+{0,1}; lane>=16: m=lane-16, k=k0+{2,3}
__device__ __forceinline__ v2f frag_m(const float* base, int row0, int ld,
                                      int k0, int lane) {
  int m = lane & 15;
  int k = k0 + ((lane >> 4) << 1);
  const float* p = base + (size_t)(row0 + m) * ld + k;
  v2f r;
  r.x = p[0];
  r.y = p[1];
  return r;
}

// B-operand fragment from K-major [K][ld] storage (row = k, col = n).
__device__ __forceinline__ v2f frag_n(const float* base, int col0, int ld,
                                      int k0, int lane) {
  int n = col0 + (lane & 15);
  int k = k0 + ((lane >> 4) << 1);
  v2f r;
  r.x = base[(size_t)k * ld + n];
  r.y = base[(size_t)(k + 1) * ld + n];
  return r;
}

// C/D fragment store: VGPR i, lanes 0-15 -> M=i, lanes 16-31 -> M=i+8.
__device__ __forceinline__ void store_c(float* base, int row0, int col0, int ld,
                                        int lane, v8f c) {
  int r0 = row0 + ((lane >> 4) << 3);
  int cc = col0 + (lane & 15);
#pragma unroll
  for (int i = 0; i < 8; ++i) base[(size_t)(r0 + i) * ld + cc] = c[i];
}

#define D_DIM 2048
#define B_Q 256
#define K_NN 64
#define NCLU 5000
#define NROWS 50000
#define NHID 512
#define H2C 256

// ---------------- 1) cluster segment-sum (atomic) ----------------
__global__ void k_cluster_accum(const float* __restrict__ feat,
                                const int* __restrict__ labels,
                                float* __restrict__ csum,
                                float* __restrict__ cnt) {
  int n = blockIdx.x;
  int lab = labels[n];
  const float4* src = (const float4*)(feat + (size_t)n * D_DIM);
  float* dst = csum + (size_t)lab * D_DIM;
  int t = threadIdx.x;
#pragma unroll
  for (int h = 0; h < 2; ++h) {
    float4 v = src[t * 2 + h];
    int o = t * 8 + h * 4;
    atomicAdd(dst + o + 0, v.x);
    atomicAdd(dst + o + 1, v.y);
    atomicAdd(dst + o + 2, v.z);
    atomicAdd(dst + o + 3, v.w);
  }
  if (t == 0) atomicAdd(cnt + lab, 1.0f);
}

// ---------------- 2) mean (in place) + simm = sum(mean^2) ----------------
__global__ void k_finalize(float* __restrict__ csum,
                           const float* __restrict__ cnt,
                           float* __restrict__ simm) {
  __shared__ float red[256];
  int c = blockIdx.x, t = threadIdx.x;
  float inv = 1.0f / fmaxf(cnt[c], 1.0f);
  float* row = csum + (size_t)c * D_DIM;
  float acc = 0.f;
#pragma unroll
  for (int i = 0; i < 8; ++i) {
    int o = t + (i << 8);
    float v = row[o] * inv;
    row[o] = v;
    acc += v * v;
  }
  red[t] = acc;
  __syncthreads();
  for (int s = 128; s > 0; s >>= 1) {
    if (t < s) red[t] += red[t + s];
    __syncthreads();
  }
  if (t == 0) simm[c] = red[0];
}

// ---------------- 3) gather clu_feat (pivot = raw feature) ----------------
__global__ void k_build_feat(const float* __restrict__ feat,
                             const float* __restrict__ cmean,
                             const int* __restrict__ indexes,
                             const int* __restrict__ labels,
                             const int* __restrict__ knn,
                             float* __restrict__ X) {
  int bk = blockIdx.x;
  int b = bk >> 6, k = bk & 63;
  const float* src = (k == 0) ? (feat + (size_t)indexes[b] * D_DIM)
                              : (cmean + (size_t)labels[knn[bk]] * D_DIM);
  float* dst = X + (size_t)bk * D_DIM;
  int t = threadIdx.x;
  const float4* s4 = (const float4*)src;
  float4* d4 = (float4*)dst;
  d4[t] = s4[t];
  d4[t + 256] = s4[t + 256];
}

// ---------------- 4) A = softmax(w_j * (F F^T)) via WMMA ----------------
__global__ void k_similarity(const float* __restrict__ X,
                             const float* __restrict__ all_pred,
                             float* __restrict__ Aout) {
  __shared__ float Fs[64 * 33];
  __shared__ float As[64 * 65];
  __shared__ float wcol[64];
  int b = blockIdx.x, t = threadIdx.x;
  int lane = t & 31, wave = t >> 5;
  if (t < 64) wcol[t] = expf(all_pred[(size_t)((b << 6) + t) * 2 + 1]);
  int t0 = wave * 2;
  int mi0 = t0 >> 2, nj0 = t0 & 3;
  int mi1 = (t0 + 1) >> 2, nj1 = (t0 + 1) & 3;
  v8f acc0 = {}, acc1 = {};
  const float* Xb = X + ((size_t)b << 6) * D_DIM;
  int lr = t >> 2, lc = (t & 3) * 8;
  for (int k0 = 0; k0 < D_DIM; k0 += 32) {
    const float4* s4 = (const float4*)(Xb + (size_t)lr * D_DIM + k0 + lc);
    float4 va = s4[0], vb = s4[1];
    float* d = Fs + lr * 33 + lc;
    d[0] = va.x; d[1] = va.y; d[2] = va.z; d[3] = va.w;
    d[4] = vb.x; d[5] = vb.y; d[6] = vb.z; d[7] = vb.w;
    __syncthreads();
#pragma unroll
    for (int kk = 0; kk < 32; kk += 4) {
      // B = F^T, so B-fragment is also a row-gather of F (rows 16*nj)
      acc0 = wmma_f32(frag_m(Fs, mi0 * 16, 33, kk, lane),
                      frag_m(Fs, nj0 * 16, 33, kk, lane), acc0);
      acc1 = wmma_f32(frag_m(Fs, mi1 * 16, 33, kk, lane),
                      frag_m(Fs, nj1 * 16, 33, kk, lane), acc1);
    }
    __syncthreads();
  }
  store_c(As, mi0 * 16, nj0 * 16, 65, lane, acc0);
  store_c(As, mi1 * 16, nj1 * 16, 65, lane, acc1);
  __syncthreads();
  if (t < 64) {
    float* row = As + t * 65;
    float m = -1e30f;
    for (int j = 0; j < 64; ++j) {
      float v = row[j] * wcol[j];
      row[j] = v;
      m = fmaxf(m, v);
    }
    float s = 0.f;
    for (int j = 0; j < 64; ++j) {
      float e = expf(row[j] - m);
      row[j] = e;
      s += e;
    }
    float is = 1.0f / s;
    float* out = Aout + (size_t)((b << 6) + t) * 64;
    for (int j = 0; j < 64; ++j) out[j] = row[j] * is;
  }
}

// ---------------- 5) row rnorms ----------------
__global__ void k_norms(const float* __restrict__ X, float* __restrict__ norms) {
  __shared__ float red[256];
  int bk = blockIdx.x, t = threadIdx.x;
  const float* row = X + (size_t)bk * D_DIM;
  float acc = 0.f;
#pragma unroll
  for (int i = 0; i < 8; ++i) {
    float v = row[t + (i << 8)];
    acc += v * v;
  }
  red[t] = acc;
  __syncthreads();
  for (int s = 128; s > 0; s >>= 1) {
    if (t < s) red[t] += red[t + s];
    __syncthreads();
  }
  if (t == 0) norms[bk] = 1.0f / sqrtf(red[0]);
}

// ---------------- 6) x = x/||x|| - pivot/||pivot|| ----------------
__global__ void k_center(const float* __restrict__ X,
                         const float* __restrict__ norms,
                         float* __restrict__ Xn) {
  int bk = blockIdx.x, t = threadIdx.x;
  int b = bk >> 6;
  float rn = norms[bk], rp = norms[b << 6];
  const float* row = X + (size_t)bk * D_DIM;
  const float* piv = X + ((size_t)(b << 6)) * D_DIM;
  float* out = Xn + (size_t)bk * D_DIM;
#pragma unroll
  for (int i = 0; i < 8; ++i) {
    int o = t + (i << 8);
    out[o] = row[o] * rn - piv[o] * rp;
  }
}

// ---------------- 7) AG = A @ Xin  (M=64, K=64, N=width) ----------------
__global__ void k_spmm(const float* __restrict__ Aadj,
                       const float* __restrict__ Xin, float* __restrict__ out,
                       int width) {
  __shared__ float As[64 * 65];
  __shared__ float Xs[64 * 65];
  int b = blockIdx.x, n0 = blockIdx.y * 64;
  int t = threadIdx.x, lane = t & 31, wave = t >> 5;
  const float* Ab = Aadj + ((size_t)b << 6) * 64;
  const float* Xb = Xin + ((size_t)b << 6) * width;
#pragma unroll
  for (int i = 0; i < 16; ++i) {
    int idx = (i << 8) + t;
    int r = idx >> 6, c = idx & 63;
    As[r * 65 + c] = Ab[idx];
    Xs[r * 65 + c] = Xb[(size_t)r * width + n0 + c];
  }
  __syncthreads();
  int t0 = wave * 2;
  int mi0 = t0 >> 2, nj0 = t0 & 3;
  int mi1 = (t0 + 1) >> 2, nj1 = (t0 + 1) & 3;
  v8f acc0 = {}, acc1 = {};
#pragma unroll
  for (int kk = 0; kk < 64; kk += 4) {
    acc0 = wmma_f32(frag_m(As, mi0 * 16, 65, kk, lane),
                    frag_n(Xs, nj0 * 16, 65, kk, lane), acc0);
    acc1 = wmma_f32(frag_m(As, mi1 * 16, 65, kk, lane),
                    frag_n(Xs, nj1 * 16, 65, kk, lane), acc1);
  }
  float* ob = out + ((size_t)b << 6) * width;
  store_c(ob, mi0 * 16, n0 + nj0 * 16, width, lane, acc0);
  store_c(ob, mi1 * 16, n0 + nj1 * 16, width, lane, acc1);
}

// ------- 8) out = relu(Xa@Wa + Xb@Wb + bias) : fused GraphConv GEMM -------
__global__ void k_dualgemm(const float* __restrict__ Xa,
                           const float* __restrict__ Xb,
                           const float* __restrict__ Wa,
                           const float* __restrict__ Wb,
                           const float* __restrict__ bias,
                           float* __restrict__ out, int Ka, int Nout) {
  __shared__ float Xas[64 * 33];
  __shared__ float Xbs[64 * 33];
  int b = blockIdx.x, n0 = blockIdx.y * 64;
  int t = threadIdx.x, lane = t & 31, wave = t >> 5;
  int t0 = wave * 2;
  int mi0 = t0 >> 2, nj0 = t0 & 3;
  int mi1 = (t0 + 1) >> 2, nj1 = (t0 + 1) & 3;
  v8f acc0 = {}, acc1 = {};
  const float* XaB = Xa + ((size_t)b << 6) * Ka;
  const float* XbB = Xb + ((size_t)b << 6) * Ka;
  int lr = t >> 2, lc = (t & 3) * 8;
  for (int k0 = 0; k0 < Ka; k0 += 32) {
    const float4* sa = (const float4*)(XaB + (size_t)lr * Ka + k0 + lc);
    const float4* sb = (const float4*)(XbB + (size_t)lr * Ka + k0 + lc);
    float4 a0 = sa[0], a1 = sa[1], b0 = sb[0], b1 = sb[1];
    float* da = Xas + lr * 33 + lc;
    float* db = Xbs + lr * 33 + lc;
    da[0] = a0.x; da[1] = a0.y; da[2] = a0.z; da[3] = a0.w;
    da[4] = a1.x; da[5] = a1.y; da[6] = a1.z; da[7] = a1.w;
    db[0] = b0.x; db[1] = b0.y; db[2] = b0.z; db[3] = b0.w;
    db[4] = b1.x; db[5] = b1.y; db[6] = b1.z; db[7] = b1.w;
    __syncthreads();
#pragma unroll
    for (int kk = 0; kk < 32; kk += 4) {
      int kg = k0 + kk;
      acc0 = wmma_f32(frag_m(Xas, mi0 * 16, 33, kk, lane),
                      frag_n(Wa, n0 + nj0 * 16, Nout, kg, lane), acc0);
      acc0 = wmma_f32(frag_m(Xbs, mi0 * 16, 33, kk, lane),
                      frag_n(Wb, n0 + nj0 * 16, Nout, kg, lane), acc0);
      acc1 = wmma_f32(frag_m(Xas, mi1 * 16, 33, kk, lane),
                      frag_n(Wa, n0 + nj1 * 16, Nout, kg, lane), acc1);
      acc1 = wmma_f32(frag_m(Xbs, mi1 * 16, 33, kk, lane),
                      frag_n(Wb, n0 + nj1 * 16, Nout, kg, lane), acc1);
    }
    __syncthreads();
  }
  float bv0 = bias[n0 + nj0 * 16 + (lane & 15)];
  float bv1 = bias[n0 + nj1 * 16 + (lane & 15)];
  v8f r0, r1;
#pragma unroll
  for (int i = 0; i < 8; ++i) {
    r0[i] = fmaxf(acc0[i] + bv0, 0.0f);
    r1[i] = fmaxf(acc1[i] + bv1, 0.0f);
  }
  float* ob = out + ((size_t)b << 6) * Nout;
  store_c(ob, mi0 * 16, n0 + nj0 * 16, Nout, lane, r0);
  store_c(ob, mi1 * 16, n0 + nj1 * 16, Nout, lane, r1);
}

// ------- 9) classifier: PReLU(x@Wc1+bc1)@Wc2+bc2 -> softmax2 -------
__global__ void k_classifier(const float* __restrict__ Xin,
                             const float* __restrict__ Wc1,
                             const float* __restrict__ bc1,
                             const float* __restrict__ pa,
                             const float* __restrict__ Wc2,
                             const float* __restrict__ bc2,
                             float* __restrict__ pred) {
  __shared__ float xs[32 * 33];
  __shared__ float hs[32 * 257];
  int r0 = blockIdx.x * 32;
  int t = threadIdx.x, lane = t & 31, wave = t >> 5;
  v8f acc[4] = {};
  int lr = t >> 3, lc = (t & 7) * 4;
  for (int k0 = 0; k0 < 256; k0 += 32) {
    float4 v = *(const float4*)(Xin + (size_t)(r0 + lr) * 256 + k0 + lc);
    float* d = xs + lr * 33 + lc;
    d[0] = v.x; d[1] = v.y; d[2] = v.z; d[3] = v.w;
    __syncthreads();
#pragma unroll
    for (int kk = 0; kk < 32; kk += 4) {
#pragma unroll
      for (int q = 0; q < 4; ++q) {
        int tt = wave * 4 + q;
        int mi = tt >> 4, nj = tt & 15;
        acc[q] = wmma_f32(frag_m(xs, mi * 16, 33, kk, lane),
                          frag_n(Wc1, nj * 16, 256, k0 + kk, lane), acc[q]);
      }
    }
    __syncthreads();
  }
#pragma unroll
  for (int q = 0; q < 4; ++q) {
    int tt = wave * 4 + q;
    int mi = tt >> 4, nj = tt & 15;
    int cc = nj * 16 + (lane & 15);
    float bb = bc1[cc], aa = pa[cc];
    v8f h;
#pragma unroll
    for (int i = 0; i < 8; ++i) {
      float v = acc[q][i] + bb;
      h[i] = (v >= 0.0f) ? v : aa * v;
    }
    store_c(hs, mi * 16, nj * 16, 257, lane, h);
  }
  __syncthreads();
  if (t < 32) {
    const float* hr = hs + t * 257;
    float l0 = bc2[0], l1 = bc2[1];
    for (int j = 0; j < 256; ++j) {
      float hv = hr[j];
      l0 += hv * Wc2[j * 2];
      l1 += hv * Wc2[j * 2 + 1];
    }
    float m = fmaxf(l0, l1);
    float e0 = expf(l0 - m), e1 = expf(l1 - m);
    float inv = 1.0f / (e0 + e1);
    pred[(size_t)(r0 + t) * 2 + 0] = e0 * inv;
    pred[(size_t)(r0 + t) * 2 + 1] = e1 * inv;
  }
}

// ---------------- launch ----------------
extern "C" void kernel_launch(void* const* d_in, const int* in_sizes, int n_in,
                              void* d_out, int out_size, void* d_ws,
                              size_t ws_size, hipStream_t stream) {
  (void)in_sizes; (void)n_in; (void)out_size; (void)ws_size;
  const int* indexes = (const int*)d_in[0];
  const float* features = (const float*)d_in[1];
  const int* labels = (const int*)d_in[2];
  const int* knn = (const int*)d_in[5];
  const float* all_pred = (const float*)d_in[6];
  const float* W1 = (const float*)d_in[7];
  const float* b1 = (const float*)d_in[8];
  const float* W2 = (const float*)d_in[9];
  const float* b2 = (const float*)d_in[10];
  const float* Wc1 = (const float*)d_in[11];
  const float* bc1 = (const float*)d_in[12];
  const float* pa = (const float*)d_in[13];
  const float* Wc2 = (const float*)d_in[14];
  const float* bc2 = (const float*)d_in[15];

  float* out = (float*)d_out;
  float* pred = out;            // [256*64*2] = 32768
  float* simm = out + 32768;    // [5000]

  float* ws = (float*)d_ws;
  float* clu = ws;                    // 10,240,000 f (reused as H1: 8,388,608 f)
  float* counts = ws + 10240000;      // 5,000 f (padded)
  float* norms = ws + 10245120;       // 16,384 f
  float* Aadj = ws + 10261504;        // 1,048,576 f
  float* X = ws + 11310080;           // 33,554,432 f (clu_feat -> AG1 -> AG2)
  float* Xn = ws + 44864512;          // 33,554,432 f (x_centered -> H2)
  float* H1 = clu;                    // alias (clu_mean dead after gather)

  // zero cluster sums + counts
  hipMemsetAsync(ws, 0, (size_t)10245120 * sizeof(float), stream);

  k_cluster_accum<<<NROWS, 256, 0, stream>>>(features, labels, clu, counts);
  k_finalize<<<NCLU, 256, 0, stream>>>(clu, counts, simm);
  k_build_feat<<<B_Q * K_NN, 256, 0, stream>>>(features, clu, indexes, labels,
                                               knn, X);
  k_similarity<<<B_Q, 256, 0, stream>>>(X, all_pred, Aadj);
  k_norms<<<B_Q * K_NN, 256, 0, stream>>>(X, norms);
  k_center<<<B_Q * K_NN, 256, 0, stream>>>(X, norms, Xn);
  // gconv1: AG1 = A@Xn (into X), H1 = relu(Xn@W1a + AG1@W1b + b1)
  k_spmm<<<dim3(B_Q, D_DIM / 64), 256, 0, stream>>>(Aadj, Xn, X, D_DIM);
  k_dualgemm<<<dim3(B_Q, NHID / 64), 256, 0, stream>>>(
      Xn, X, W1, W1 + (size_t)2048 * NHID, b1, H1, D_DIM, NHID);
  // gconv2: AG2 = A@H1 (into X), H2 = relu(H1@W2a + AG2@W2b + b2) (into Xn)
  k_spmm<<<dim3(B_Q, NHID / 64), 256, 0, stream>>>(Aadj, H1, X, NHID);
  k_dualgemm<<<dim3(B_Q, H2C / 64), 256, 0, stream>>>(
      H1, X, W2, W2 + (size_t)512 * H2C, b2, Xn, NHID, H2C);
  // classifier + softmax
  k_classifier<<<B_Q * K_NN / 32, 256, 0, stream>>>(Xn, Wc1, bc1, pa, Wc2, bc2,
                                                    pred);
}